// CrossTalk_18708877541831
// MI455X (gfx1250) — compile-verified
//
#include <hip/hip_runtime.h>
#include <cstdint>

#define N_TILES  12
#define N_FIBRES 600
#define N_POS    (N_TILES * N_FIBRES)   // 7200 fibre positions
#define N_VEC4   (N_POS / 4)            // 1800 vec4 groups
#define BLOCK    256                    // 8 waves (wave32)

// Native vector types (usable with nontemporal builtins, swizzle accessors).
typedef float v4f __attribute__((ext_vector_type(4)));
typedef int   v4i __attribute__((ext_vector_type(4)));

// Exact pointee types for the gfx1250 async-to-LDS builtins:
//   b32 : AS1 int*        src, AS3 int*        dst   (probe-confirmed)
//   b128: AS1 gcc-v4i*    src, AS3 gcc-v4i*    dst   (per hipcc diagnostic)
typedef int gccv4i __attribute__((vector_size(16)));
typedef __attribute__((address_space(1))) int    gbl_i32_t;
typedef __attribute__((address_space(3))) int    lds_i32_t;
typedef __attribute__((address_space(1))) gccv4i gbl_v4i_t;
typedef __attribute__((address_space(3))) gccv4i lds_v4i_t;

// ---------------------------------------------------------------------------
// Kernel 1: pack per-position segment id + fibre-boundary flags into one int.
//   code = (seg << 2) | (hasRight << 1) | hasLeft, seg = tile*600 + fib
// ---------------------------------------------------------------------------
__global__ void CrossTalk_seg_precompute(const int* __restrict__ tile_idx,
                                         const int* __restrict__ fib_idx,
                                         int* __restrict__ seg_code) {
    int p = blockIdx.x * blockDim.x + threadIdx.x;
    if (p < N_POS) {
        int f = fib_idx[p];
        int s = tile_idx[p] * N_FIBRES + f;
        seg_code[p] = (s << 2) | ((f < N_FIBRES - 1) ? 2 : 0) | ((f > 0) ? 1 : 0);
    }
}

// ---------------------------------------------------------------------------
// Kernel 2: one block per wavelength channel.
//   Phase A: async DMA flux row -> LDS (b128 per lane), overlap with zeroing
//   Phase B: LDS atomic scatter-add into (tile,fibre) bins (ds_add_f32)
//   Phase C: fused tridiagonal mix + gather, b128 non-temporal stores
// ---------------------------------------------------------------------------
__global__ __launch_bounds__(BLOCK) void CrossTalk_main(
    const float* __restrict__ flux,
    const float* __restrict__ eta_p,
    const int*   __restrict__ seg_code,
    float*       __restrict__ out) {

    __shared__ float lds_flux[N_POS];   // 28.8 KB staged flux row
    __shared__ float lds_bin[N_POS];    // 28.8 KB (tile,fibre) bins

    const int tid = threadIdx.x;
    const float* row = flux + (size_t)blockIdx.x * N_POS;

    // ---- Phase A: async global -> LDS DMA of the flux row (ASYNCcnt) -------
#if defined(__gfx1250__) && __has_builtin(__builtin_amdgcn_global_load_async_to_lds_b128)
    for (int i = tid; i < N_VEC4; i += BLOCK) {
        __builtin_amdgcn_global_load_async_to_lds_b128(
            (gbl_v4i_t*)(row + 4 * i), (lds_v4i_t*)&lds_flux[4 * i],
            /*offset=*/0, /*cpol=*/0);
    }
#define CT_ASYNC_USED 1
#elif defined(__gfx1250__) && __has_builtin(__builtin_amdgcn_global_load_async_to_lds_b32)
    for (int p = tid; p < N_POS; p += BLOCK) {
        __builtin_amdgcn_global_load_async_to_lds_b32(
            (gbl_i32_t*)(row + p), (lds_i32_t*)&lds_flux[p],
            /*offset=*/0, /*cpol=*/0);
    }
#define CT_ASYNC_USED 1
#else
    for (int p = tid; p < N_POS; p += BLOCK) lds_flux[p] = row[p];
#endif

    // Overlap: zero the bins (ds_store_b128) while the DMA is in flight.
    {
        v4f* bin4 = (v4f*)lds_bin;
        const v4f z = {0.f, 0.f, 0.f, 0.f};
        for (int i = tid; i < N_VEC4; i += BLOCK) bin4[i] = z;
    }

#ifdef CT_ASYNC_USED
#if __has_builtin(__builtin_amdgcn_s_wait_asynccnt)
    __builtin_amdgcn_s_wait_asynccnt(0);
#else
    asm volatile("s_wait_asynccnt 0" ::: "memory");
#endif
#endif
    __syncthreads();

    // ---- Phase B: scatter-add into bins ------------------------------------
    {
        const v4i* seg4  = (const v4i*)seg_code;       // L2-resident, 28.8 KB
        const v4f* flux4 = (const v4f*)lds_flux;
        for (int i = tid; i < N_VEC4; i += BLOCK) {
            v4i c = seg4[i];
            v4f v = flux4[i];
            atomicAdd(&lds_bin[c.x >> 2], v.x);
            atomicAdd(&lds_bin[c.y >> 2], v.y);
            atomicAdd(&lds_bin[c.z >> 2], v.z);
            atomicAdd(&lds_bin[c.w >> 2], v.w);
        }
    }
    __syncthreads();

    // ---- Phase C: fused tridiagonal mix + gather ---------------------------
    const float eta = eta_p[0];
    const float c0  = 1.0f - 2.0f * eta;
    {
        const v4i* seg4 = (const v4i*)seg_code;
        v4f* out4 = (v4f*)(out + (size_t)blockIdx.x * N_POS);
        for (int i = tid; i < N_VEC4; i += BLOCK) {
            v4i c = seg4[i];
            v4f r;
            {
                int s = c.x >> 2; float v = c0 * lds_bin[s];
                if (c.x & 1) v += eta * lds_bin[s - 1];
                if (c.x & 2) v += eta * lds_bin[s + 1];
                r.x = v;
            }
            {
                int s = c.y >> 2; float v = c0 * lds_bin[s];
                if (c.y & 1) v += eta * lds_bin[s - 1];
                if (c.y & 2) v += eta * lds_bin[s + 1];
                r.y = v;
            }
            {
                int s = c.z >> 2; float v = c0 * lds_bin[s];
                if (c.z & 1) v += eta * lds_bin[s - 1];
                if (c.z & 2) v += eta * lds_bin[s + 1];
                r.z = v;
            }
            {
                int s = c.w >> 2; float v = c0 * lds_bin[s];
                if (c.w & 1) v += eta * lds_bin[s - 1];
                if (c.w & 2) v += eta * lds_bin[s + 1];
                r.w = v;
            }
            __builtin_nontemporal_store(r, &out4[i]);  // output never re-read
        }
    }
}

// ---------------------------------------------------------------------------
extern "C" void kernel_launch(void* const* d_in, const int* in_sizes, int n_in,
                              void* d_out, int out_size, void* d_ws, size_t ws_size,
                              hipStream_t stream) {
    const float* flux     = (const float*)d_in[0];   // (N_WAVE, 7200) f32
    const float* eta      = (const float*)d_in[1];   // scalar f32
    const int*   tile_idx = (const int*)d_in[2];     // (7200,) int
    const int*   fib_idx  = (const int*)d_in[3];     // (7200,) int
    float*       out      = (float*)d_out;
    int*         seg_code = (int*)d_ws;              // 7200 ints of scratch

    const int n_wave = in_sizes[0] / N_POS;          // 4096

    CrossTalk_seg_precompute<<<(N_POS + BLOCK - 1) / BLOCK, BLOCK, 0, stream>>>(
        tile_idx, fib_idx, seg_code);

    CrossTalk_main<<<n_wave, BLOCK, 0, stream>>>(flux, eta, seg_code, out);
}